// VectorDB_72447508349188
// MI455X (gfx1250) — compile-verified
//
#include <hip/hip_runtime.h>
#include <float.h>
#include <math.h>

// ---------------------------------------------------------------------------
// Cosine-similarity vector search, MI455X (gfx1250, wave32).
//   query[384] f32, db[1000000][384] f32, k=10
// Bandwidth-bound: 1.536 GB / 23.3 TB/s ~ 66 us floor. Strategy:
//   * stream db once, f32 end-to-end (precision == reference)
//   * V_WMMA_F32_16X16X4_F32 computes 16 dot products per wave-tile
//   * same A fragment fed as B gives Gram matrix -> row norms for free
//   * hierarchical top-k: per-block wave argmax -> single-block merge
// ---------------------------------------------------------------------------

#define DIMS            384
#define TILE_M          16
#define WAVES_PER_BLK   8
#define TILES_PER_WAVE  8
#define TILES_PER_BLK   (WAVES_PER_BLK * TILES_PER_WAVE)   /* 64  */
#define ROWS_PER_BLK    (TILES_PER_BLK * TILE_M)           /* 1024 */
#define KMAX            32

typedef __attribute__((ext_vector_type(2))) float v2f;
typedef __attribute__((ext_vector_type(8))) float v8f;

// ---------------------------------------------------------------------------
// Stage 0: qn = q / ||q||   (1 block, 128 threads)
// ---------------------------------------------------------------------------
__global__ __launch_bounds__(128)
void vdb_qnorm_kernel(const float* __restrict__ q, float* __restrict__ qn) {
    __shared__ float red[128];
    const int tid = threadIdx.x;
    float ss = 0.f;
    for (int i = tid; i < DIMS; i += 128) { float v = q[i]; ss += v * v; }
    red[tid] = ss;
    __syncthreads();
    for (int s = 64; s >= 1; s >>= 1) {
        if (tid < s) red[tid] += red[tid + s];
        __syncthreads();
    }
    const float nrm = sqrtf(red[0]);
    for (int i = tid; i < DIMS; i += 128) qn[i] = q[i] / nrm;
}

// ---------------------------------------------------------------------------
// Stage 1: scores via WMMA f32 16x16x4 + per-block top-K candidates
// ---------------------------------------------------------------------------
__global__ __launch_bounds__(256)
void vdb_score_topk_kernel(const float* __restrict__ db,
                           const float* __restrict__ qn,
                           float* __restrict__ candScore,
                           int*   __restrict__ candIdx,
                           int N, int K)
{
    __shared__ float qLDS[DIMS];
    __shared__ float scoreLDS[ROWS_PER_BLK];

    const int tid  = threadIdx.x;
    const int lane = tid & 31;
    const int wave = tid >> 5;

    for (int i = tid; i < DIMS; i += 256)         qLDS[i]     = qn[i];
    for (int i = tid; i < ROWS_PER_BLK; i += 256) scoreLDS[i] = -FLT_MAX;
    __syncthreads();

    const int       ntiles     = (N + TILE_M - 1) / TILE_M;
    const int       blockTile0 = blockIdx.x * TILES_PER_BLK;
    const long long blockRow0  = (long long)blockTile0 * TILE_M;

    // kh selects which half of the K=4 slice this lane supplies:
    // lanes 0-15 -> K={k0,k0+1}, lanes 16-31 -> K={k0+2,k0+3}  (A layout),
    // and identically K=v / K=2+v for the B (4x16) layout.
    const int kh = (lane >> 4) << 1;

    for (int i = 0; i < TILES_PER_WAVE; ++i) {
        const int tLocal = i * WAVES_PER_BLK + wave;   // concurrent waves hit adjacent tiles
        const int tile   = blockTile0 + tLocal;
        if (tile >= ntiles) break;                     // wave-uniform

        const long long row0 = (long long)tile * TILE_M;
        const long long row  = row0 + (lane & 15);
        const long long rowC = (row < N) ? row : (long long)(N - 1);
        const float* ap = db + rowC * DIMS + kh;

        v8f accD = {};   // accD[r], lane n<16: C[r][n]   ; lane n>=16: C[r+8][n-16]
        v8f accN = {};   // Gram matrix accumulate: diag = sum of squares per row

        #pragma unroll 4
        for (int k0 = 0; k0 < DIMS; k0 += 4) {
            v2f a = *(const v2f*)(ap + k0);            // A fragment (16x4 f32 tile)
            v2f b = *(const v2f*)(&qLDS[k0 + kh]);     // B fragment = query bcast all 16 cols
            accD = __builtin_amdgcn_wmma_f32_16x16x4_f32(
                       false, a, false, b, (short)0, accD, false, false);
            // A fragment layout == (A^T as B) layout -> A x A^T, diag = ||row||^2
            accN = __builtin_amdgcn_wmma_f32_16x16x4_f32(
                       false, a, false, a, (short)0, accN, false, false);
        }

        // Extract: dot(row r)  at (M=r,N=0)   -> VGPR r, lane 0 / lane 16
        //          ||r||^2     at (M=r,N=r)   -> VGPR r%8, lane r (r<8) or lane r+16 (r>=8)
        const int sbase = tLocal * TILE_M;
        #pragma unroll
        for (int r = 0; r < 8; ++r) {
            const float dlo = __shfl(accD[r], 0, 32);
            const float dhi = __shfl(accD[r], 16, 32);
            const float nlo = __shfl(accN[r], r, 32);        // (M=r,    N=r)
            const float nhi = __shfl(accN[r], r + 24, 32);   // (M=r+8,  N=r+8)
            if (lane == 0) {
                scoreLDS[sbase + r] = (row0 + r < N)
                    ? dlo / fmaxf(sqrtf(nlo), 1e-12f) : -FLT_MAX;
            }
            if (lane == 1) {
                scoreLDS[sbase + 8 + r] = (row0 + 8 + r < N)
                    ? dhi / fmaxf(sqrtf(nhi), 1e-12f) : -FLT_MAX;
            }
        }
    }
    __syncthreads();

    // Wave 0: iterative argmax over the block's 1024 scores, fully in registers
    // (no barriers inside the K loop).
    if (wave == 0) {
        float ls[32];
        #pragma unroll
        for (int c = 0; c < 32; ++c) ls[c] = scoreLDS[c * 32 + lane];

        for (int it = 0; it < K; ++it) {
            float best = -FLT_MAX; int bi = -1;
            #pragma unroll
            for (int c = 0; c < 32; ++c)
                if (ls[c] > best) { best = ls[c]; bi = c * 32 + lane; }

            #pragma unroll
            for (int off = 16; off >= 1; off >>= 1) {
                const float ob  = __shfl_xor(best, off, 32);
                const int   obi = __shfl_xor(bi,   off, 32);
                if (ob > best || (ob == best && (unsigned)obi < (unsigned)bi)) {
                    best = ob; bi = obi;
                }
            }
            if (lane == 0) {
                candScore[blockIdx.x * K + it] = best;
                candIdx  [blockIdx.x * K + it] =
                    (int)(blockRow0 + (bi < 0 ? 0 : bi));
            }
            #pragma unroll
            for (int c = 0; c < 32; ++c)
                if (c * 32 + lane == bi) ls[c] = -FLT_MAX;
        }
    }
}

// ---------------------------------------------------------------------------
// Stage 2: merge nBlocks*K candidates -> global top-K (1 block, 256 threads)
// ---------------------------------------------------------------------------
__global__ __launch_bounds__(256)
void vdb_merge_kernel(const float* __restrict__ candScore,
                      const int*   __restrict__ candIdx,
                      int C, int K,
                      float* __restrict__ outScore,
                      int*   __restrict__ outIdx)
{
    __shared__ float rs[256];
    __shared__ int   ri[256];
    __shared__ int   chosen[KMAX];
    const int tid = threadIdx.x;

    for (int it = 0; it < K; ++it) {
        float best = -FLT_MAX; int bi = -1;
        for (int j = tid; j < C; j += 256) {
            bool skip = false;
            for (int t = 0; t < it; ++t) skip |= (chosen[t] == j);
            const float s = candScore[j];
            if (!skip && (s > best || (s == best && (unsigned)j < (unsigned)bi))) {
                best = s; bi = j;
            }
        }
        rs[tid] = best; ri[tid] = bi;
        __syncthreads();
        for (int s = 128; s >= 1; s >>= 1) {
            if (tid < s) {
                const float os = rs[tid + s]; const int oi = ri[tid + s];
                if (os > rs[tid] || (os == rs[tid] && (unsigned)oi < (unsigned)ri[tid])) {
                    rs[tid] = os; ri[tid] = oi;
                }
            }
            __syncthreads();
        }
        if (tid == 0) {
            const int slot = (ri[0] < 0) ? 0 : ri[0];
            chosen[it]   = slot;
            outScore[it] = rs[0];
            outIdx[it]   = candIdx[slot];
        }
        __syncthreads();
    }
}

// ---------------------------------------------------------------------------
extern "C" void kernel_launch(void* const* d_in, const int* in_sizes, int n_in,
                              void* d_out, int out_size, void* d_ws, size_t ws_size,
                              hipStream_t stream) {
    const float* q  = (const float*)d_in[0];
    const float* db = (const float*)d_in[1];
    // d_in[2] is k, but it lives on device; recover it from out_size = 2*k.
    const int N = in_sizes[1] / DIMS;
    int K = out_size / 2;
    if (K < 1) K = 1;
    if (K > KMAX) K = KMAX;

    const int ntiles  = (N + TILE_M - 1) / TILE_M;
    const int nBlocks = (ntiles + TILES_PER_BLK - 1) / TILES_PER_BLK;
    const int C       = nBlocks * K;

    // ws layout (floats): [0,512) qn | [512, 512+C) candScore | [512+C, 512+2C) candIdx
    float* wsf       = (float*)d_ws;
    float* qn        = wsf;
    float* candScore = wsf + 512;
    int*   candIdx   = (int*)(wsf + 512 + C);

    float* outScore = (float*)d_out;
    int*   outIdx   = (int*)d_out + K;

    vdb_qnorm_kernel<<<1, 128, 0, stream>>>(q, qn);
    vdb_score_topk_kernel<<<nBlocks, 256, 0, stream>>>(db, qn, candScore, candIdx, N, K);
    vdb_merge_kernel<<<1, 256, 0, stream>>>(candScore, candIdx, C, K, outScore, outIdx);
}